// Algebraic_65970697666729
// MI455X (gfx1250) — compile-verified
//
#include <hip/hip_runtime.h>
#include <stdint.h>
#include <stddef.h>

// CDNA5 / gfx1250, wave32.
typedef __attribute__((ext_vector_type(2))) float v2f;
typedef __attribute__((ext_vector_type(8))) float v8f;

#define NFEAT 16
#define NCOL  696          // 16 + 120 + 560
#define RPB   16           // rows per block
#define THREADS 256
#define QPR   (NCOL / 4)   // 174 float4 groups per row

// Column table matching itertools.combinations order:
//   cols [0,16):   identity (handled by c<16 branch)
//   cols [16,136): pairs (i<j lexicographic)  -> low 8 bits = i*16+j, high = 16 (the "1.0" slot)
//   cols [136,696): triples (i<j<k lexicographic) -> low 8 bits = i*16+j, high = k
struct ColTab { uint16_t v[NCOL]; };
static constexpr ColTab make_tab() {
    ColTab t{};
    int c = 0;
    for (int i = 0; i < NFEAT; ++i) t.v[c++] = 0;
    for (int i = 0; i < NFEAT; ++i)
        for (int j = i + 1; j < NFEAT; ++j)
            t.v[c++] = (uint16_t)((i * 16 + j) | (16u << 8));
    for (int i = 0; i < NFEAT; ++i)
        for (int j = i + 1; j < NFEAT; ++j)
            for (int k = j + 1; k < NFEAT; ++k)
                t.v[c++] = (uint16_t)((i * 16 + j) | ((unsigned)k << 8));
    return t;
}
__constant__ ColTab g_tab = make_tab();

__global__ __launch_bounds__(THREADS)
void Algebraic_65970697666729_kernel(const float* __restrict__ x,
                                     float* __restrict__ out,
                                     int rows) {
    __shared__ float xt[RPB][NFEAT + 1];   // +1 = constant 1.0 slot
    __shared__ float P[RPB][256];          // per-row 16x16 outer-product matrix

    const int tid  = threadIdx.x;
    const int row0 = blockIdx.x * RPB;

    // ---- Stage 16x16 f32 tile into LDS via CDNA5 async global->LDS DMA ----
    {
        const int r = tid >> 4;
        const int c = tid & 15;
        const int grow = row0 + r;
        if (grow < rows) {
            unsigned lds_off = (unsigned)(uintptr_t)&xt[r][c];      // low 32 bits of generic ptr = LDS offset
            const float* gp = x + (size_t)grow * NFEAT + c;
            asm volatile("global_load_async_to_lds_b32 %0, %1, off"
                         :: "v"(lds_off), "v"(gp)
                         : "memory");
        } else {
            xt[r][c] = 0.0f;
        }
        if (tid < RPB) xt[tid][NFEAT] = 1.0f;
        asm volatile("s_wait_asynccnt 0x0" ::: "memory");
    }
    __syncthreads();

    // ---- Pair products: per-row outer product x (x)T x via V_WMMA_F32_16X16X4_F32 ----
    // A (16x4): only K=0 column nonzero -> VGPR0 lanes 0-15 = x[m], all else 0.
    // B (4x16): only K=0 row nonzero    -> VGPR0 lanes 0-15 = x[n], all else 0.
    // Same register image for A and B.  D[m,n] = x[m]*x[n], exact in f32.
    {
        const int wave = tid >> 5;
        const int lane = tid & 31;
#pragma unroll
        for (int rr = 0; rr < 2; ++rr) {
            const int r = wave * 2 + rr;
            const float xv = (lane < 16) ? xt[r][lane] : 0.0f;
            v2f a;
            a[0] = xv;
            a[1] = 0.0f;
            v8f acc = {};
            v8f d = __builtin_amdgcn_wmma_f32_16x16x4_f32(
                false, a, false, a, (short)0, acc, false, false);
            // C/D layout: VGPR v, lane L -> M = v + 8*(L>=16), N = L&15
            const int m0 = (lane >> 4) * 8;
            const int n  = lane & 15;
#pragma unroll
            for (int v = 0; v < 8; ++v)
                P[r][(m0 + v) * 16 + n] = d[v];
        }
    }
    __syncthreads();

    // ---- Stream outputs: coalesced float4 (b128) stores, 16*696 elements/block ----
    for (int q = tid; q < RPB * QPR; q += THREADS) {
        const int r    = q / QPR;
        const int c0   = (q - r * QPR) * 4;
        const int grow = row0 + r;
        if (grow >= rows) continue;
        float vals[4];
#pragma unroll
        for (int u = 0; u < 4; ++u) {
            const int c = c0 + u;
            float val;
            if (c < NFEAT) {
                val = xt[r][c];                       // identity columns
            } else {
                const unsigned e = g_tab.v[c];
                val = P[r][e & 255u] * xt[r][e >> 8]; // pair (x1.0) or triple
            }
            vals[u] = val;
        }
        float4 o = make_float4(vals[0], vals[1], vals[2], vals[3]);
        *(float4*)(out + (size_t)grow * NCOL + c0) = o;   // 16B-aligned: 696 % 4 == 0
    }
}

extern "C" void kernel_launch(void* const* d_in, const int* in_sizes, int n_in,
                              void* d_out, int out_size, void* d_ws, size_t ws_size,
                              hipStream_t stream) {
    (void)n_in; (void)out_size; (void)d_ws; (void)ws_size;
    const float* x = (const float*)d_in[0];
    float* out = (float*)d_out;
    const int rows = in_sizes[0] / NFEAT;              // 131072
    const int blocks = (rows + RPB - 1) / RPB;         // 8192
    Algebraic_65970697666729_kernel<<<blocks, THREADS, 0, stream>>>(x, out, rows);
}